// CARAFE_31155692765210
// MI455X (gfx1250) — compile-verified
//
#include <hip/hip_runtime.h>
#include <hip/hip_bf16.h>

typedef __attribute__((ext_vector_type(2))) float v2f;
typedef __attribute__((ext_vector_type(8))) float v8f;
typedef unsigned int uint;
typedef uint u32x4 __attribute__((ext_vector_type(4)));
typedef int  i32x8 __attribute__((ext_vector_type(8)));
typedef int  i32x4 __attribute__((ext_vector_type(4)));

#ifndef __has_builtin
#define __has_builtin(x) 0
#endif
#if __has_builtin(__builtin_amdgcn_tensor_load_to_lds) && defined(__gfx1250__)
#define HAVE_TDM 1
#else
#define HAVE_TDM 0
#endif

#define BB    8
#define CI    256
#define CCH   64
#define HH    64
#define WWD   64
#define HWN   4096              // 64*64
#define NPOS  32768             // BB*HWN
#define ENCCH 36                // S^2*K^2
#define KE    576               // CCH * 9

// workspace layout (in floats)
#define OFS_COMP 0
#define OFS_ENC  (BB*CCH*HWN)                 // 2,097,152
#define OFS_WSM  (OFS_ENC + BB*ENCCH*HWN)     // 3,276,800
#define OFS_STAT (OFS_WSM + BB*ENCCH*HWN)     // 4,456,448
#define OFS_W2P  (OFS_STAT + 256)             // padded k-major w2: 576*48
// stats: sum[64] | sumsq[64] | scale[64] | shift[64]

// ---------------------------------------------------------------------------
// Tensor Data Mover helper: up-to-3D tiled load Global -> LDS.
// D# packed per CDNA5 ISA ch.8: group0 {count,lds,global,type}, group1 dims.
// X dim is contiguous (stride 1); Y stride = stride0; Z stride = stride1
// (all in 4-byte elements, data_size = 4B). This toolchain's builtin takes
// 6 operands: groups 0..3 plus a trailing 8-dword vector (zeroed: unused
// descriptor state for <=3D tiles) and the cache-policy immediate.
// ---------------------------------------------------------------------------
#if HAVE_TDM
__device__ inline void tdm_load_to_lds_f32(uint ldsOff, const void* gptr,
                                           uint tile0, uint tile1, uint tile2,
                                           uint tdim0, uint tdim1, uint tdim2,
                                           unsigned long long stride0,
                                           unsigned long long stride1) {
  const unsigned long long ga = (unsigned long long)(size_t)gptr;
  u32x4 g0;
  g0[0] = 1u;                                            // count=1, user D#
  g0[1] = ldsOff;                                        // lds_addr (bytes)
  g0[2] = (uint)(ga & 0xFFFFFFFFull);                    // global_addr[31:0]
  g0[3] = (uint)((ga >> 32) & 0x01FFFFFFu) | (2u << 30); // addr[56:32]|type=2
  i32x8 g1;
  g1[0] = (int)(2u << 16);                               // data_size=4B
  g1[1] = (int)(tdim0 << 16);                            // tensor_dim0[15:0]
  g1[2] = (int)((tdim0 >> 16) | (tdim1 << 16));          // dim0[31:16]|dim1[15:0]
  g1[3] = (int)((tdim1 >> 16) | (tile0 << 16));          // dim1[31:16]|tile0
  g1[4] = (int)(tile1 | (tile2 << 16));                  // tile1 | tile2
  g1[5] = (int)(stride0 & 0xFFFFFFFFull);                // dim0_stride[31:0]
  g1[6] = (int)(((stride0 >> 32) & 0xFFFFull) |
                ((uint)(stride1 & 0xFFFFull) << 16));    // |dim1_stride[15:0]
  g1[7] = (int)((stride1 >> 16) & 0xFFFFFFFFull);        // dim1_stride[47:16]
  i32x4 g2;
  g2[0] = (int)tdim2; g2[1] = 0; g2[2] = 0; g2[3] = 0;   // tensor_dim2; rest 0
  i32x4 g3 = {0, 0, 0, 0};
  i32x8 g4 = {0, 0, 0, 0, 0, 0, 0, 0};                   // unused (zero)
  __builtin_amdgcn_tensor_load_to_lds(g0, g1, g2, g3, g4, 0);
}
#endif

// ---------------------------------------------------------------------------
// K1: 1x1 conv as WMMA GEMM (M=32768 positions, K=256, N=64), fused BN stats.
//     w1 staged to LDS by one TDM transposing walk: LDS layout [pair][o][2]
//     so B fragments are conflict-free ds_load_b64.
// ---------------------------------------------------------------------------
__global__ __launch_bounds__(256)
void k1_compress(const float* __restrict__ x, const float* __restrict__ w1,
                 const float* __restrict__ b1, float* __restrict__ comp,
                 float* __restrict__ stats /* sum[64], sumsq[64] */) {
  __shared__ float w1lds[CCH * CI];          // exactly 64 KB
  const int tid  = threadIdx.x;
  const int lane = tid & 31;
  const int wave = tid >> 5;

#if HAVE_TDM
  if (wave == 0) {
    // element (d,o,p) <- w1[o*256 + 2p + d]  (X=d:2, Y=o:64 s=256, Z=p:128 s=2)
    tdm_load_to_lds_f32((uint)(size_t)(void*)w1lds, w1,
                        2u, 64u, 128u, 2u, 64u, 128u, 256ull, 2ull);
    __builtin_amdgcn_s_wait_tensorcnt(0);
  }
#else
  for (int i = tid; i < CCH * CI / 2; i += 256) {   // 8192 pairs
    const int p = i >> 6, o = i & 63;
    v2f v;
    v.x = w1[o * CI + 2 * p];
    v.y = w1[o * CI + 2 * p + 1];
    *(v2f*)&w1lds[p * 128 + o * 2] = v;
  }
#endif
  __syncthreads();

  const int posBase = (blockIdx.x * 8 + wave) * 16;
  const int m   = lane & 15;
  const int khi = (lane >> 4) << 1;               // 0 or 2
  const int n   = posBase + m;
  const int b   = n >> 12;
  const int hw  = n & 4095;
  const float* xa = x + (size_t)b * CI * HWN + hw;

  v8f acc[4] = {v8f{0.f}, v8f{0.f}, v8f{0.f}, v8f{0.f}};

  for (int k = 0; k < CI; k += 4) {
    if ((k & 31) == 0)
      __builtin_prefetch(xa + (size_t)(k + 32) * HWN, 0, 1);
    const int klo = k + khi;
    v2f a;
    a.x = xa[(size_t)klo * HWN];
    a.y = xa[(size_t)(klo + 1) * HWN];
    const v2f* bl = (const v2f*)&w1lds[(klo >> 1) * 128];
#pragma unroll
    for (int nt = 0; nt < 4; ++nt) {
      acc[nt] = __builtin_amdgcn_wmma_f32_16x16x4_f32(
          false, a, false, bl[nt * 16 + m], (short)0, acc[nt], false, false);
    }
  }

  __syncthreads();                       // done reading w1lds: reuse as scratch
  float* red = w1lds;
  if (tid < 128) red[tid] = 0.f;
  __syncthreads();

#pragma unroll
  for (int nt = 0; nt < 4; ++nt) {
    const int o    = nt * 16 + m;
    const int mOff = (lane >> 4) << 3;            // 0 or 8
    const float bias = b1[o];
    float ls = 0.f, lq = 0.f;
#pragma unroll
    for (int r = 0; r < 8; ++r) {
      const int nn  = posBase + mOff + r;
      const int bb_ = nn >> 12;
      const int hw_ = nn & 4095;
      const float v = acc[nt][r] + bias;
      comp[((size_t)bb_ * CCH + o) * HWN + hw_] = v;
      ls += v;
      lq += v * v;
    }
    atomicAdd(&red[o], ls);                       // ds_add_f32
    atomicAdd(&red[64 + o], lq);
  }
  __syncthreads();
  if (tid < 128) atomicAdd(&stats[tid], red[tid]);  // global_atomic_add_f32
}

// ---------------------------------------------------------------------------
// K2: fold batch stats + gamma/beta into per-channel scale/shift
// ---------------------------------------------------------------------------
__global__ void k2_stats(const float* __restrict__ stats,
                         const float* __restrict__ gamma,
                         const float* __restrict__ beta,
                         float* __restrict__ scsh) {
  const int o = threadIdx.x;
  if (o < 64) {
    const float inv = 1.f / (float)NPOS;
    const float mu  = stats[o] * inv;
    const float var = stats[64 + o] * inv - mu * mu;  // biased variance
    const float rstd = rsqrtf(var + 1e-5f);
    const float sc = gamma[o] * rstd;
    scsh[o] = sc;
    scsh[64 + o] = beta[o] - mu * sc;
  }
}

// K2b: apply BN affine + ReLU in place on comp (keeps K3's hot loop lean)
__global__ __launch_bounds__(256)
void k2b_norm(float* __restrict__ comp, const float* __restrict__ scsh) {
  const int i = blockIdx.x * 256 + threadIdx.x;     // 2,097,152 exact
  const int o = (i >> 12) & 63;
  const float v = comp[i] * scsh[o] + scsh[64 + o];
  comp[i] = v > 0.f ? v : 0.f;
}

// K2w: pre-transpose w2 into zero-padded pair-interleaved k-major layout:
//      w2p[(kk>>1)*96 + o*2 + (kk&1)] = w2[o*576 + c*9 + t],  kk = t*64 + c
__global__ __launch_bounds__(256)
void k2w_transpose(const float* __restrict__ w2, float* __restrict__ w2p) {
  const int i = blockIdx.x * 256 + threadIdx.x;     // 576*48 = 27648 exact
  const int o  = i % 48;
  const int kk = i / 48;
  const int c  = kk & 63;
  const int t  = kk >> 6;
  const float v = (o < ENCCH) ? w2[o * KE + c * 9 + t] : 0.f;
  w2p[(kk >> 1) * 96 + o * 2 + (kk & 1)] = v;
}

// ---------------------------------------------------------------------------
// K3: 3x3 conv (64 -> 36->48pad) WMMA GEMM, tap-major K order: per tap the
//     bounds mask/base pointer are loop-invariant; B fragments are single b64.
// ---------------------------------------------------------------------------
__global__ __launch_bounds__(256)
void k3_encoder(const float* __restrict__ compn, const float* __restrict__ w2p,
                const float* __restrict__ b2, float* __restrict__ enc) {
  const int tid  = threadIdx.x;
  const int lane = tid & 31;
  const int wave = tid >> 5;
  const int posBase = (blockIdx.x * 8 + wave) * 16;
  const int m   = lane & 15;
  const int khi = (lane >> 4) << 1;
  const int n   = posBase + m;
  const int b   = n >> 12;
  const int hw  = n & 4095;
  const int h   = hw >> 6;
  const int w   = hw & 63;
  const float* cb  = compn + (size_t)b * CCH * HWN;
  const v2f*  w2v = (const v2f*)w2p;

  v8f acc[3] = {v8f{0.f}, v8f{0.f}, v8f{0.f}};

#pragma unroll
  for (int t = 0; t < 9; ++t) {
    const int dy = t / 3 - 1;
    const int dx = t - 3 * (t / 3) - 1;
    const int hh = h + dy;
    const int ww = w + dx;
    const bool valid = (hh >= 0) & (hh < 64) & (ww >= 0) & (ww < 64);
    const float* ab = cb + (hh << 6) + ww;          // + c*HWN
#pragma unroll 4
    for (int c4 = 0; c4 < 64; c4 += 4) {
      const int c0 = c4 + khi;
      v2f a;
      a.x = valid ? ab[(size_t)c0 * HWN] : 0.f;
      a.y = valid ? ab[(size_t)(c0 + 1) * HWN] : 0.f;
      const int P = t * 32 + (c0 >> 1);             // k-pair index
#pragma unroll
      for (int nt = 0; nt < 3; ++nt) {
        acc[nt] = __builtin_amdgcn_wmma_f32_16x16x4_f32(
            false, a, false, w2v[P * 48 + nt * 16 + m],
            (short)0, acc[nt], false, false);
      }
    }
  }

#pragma unroll
  for (int nt = 0; nt < 3; ++nt) {
    const int o = nt * 16 + m;
    if (o < ENCCH) {
      const float bias = b2[o];
      const int mOff = (lane >> 4) << 3;
#pragma unroll
      for (int r = 0; r < 8; ++r) {
        const int nn  = posBase + mOff + r;
        const int bb_ = nn >> 12;
        const int hw_ = nn & 4095;
        enc[((size_t)bb_ * ENCCH + o) * HWN + hw_] = acc[nt][r] + bias;
      }
    }
  }
}

// ---------------------------------------------------------------------------
// K4: softmax along the H axis (faithful to reference: softmax dim=1 -> H)
// ---------------------------------------------------------------------------
__global__ __launch_bounds__(256)
void k4_softmax(const float* __restrict__ enc, float* __restrict__ wsm) {
  const int idx = blockIdx.x * blockDim.x + threadIdx.x;
  if (idx >= BB * ENCCH * WWD) return;
  const int w   = idx & 63;
  const int bch = idx >> 6;                       // b*36 + ch
  const float* col = enc + (size_t)bch * HWN + w; // stride 64 over h
  float mx = -3.4e38f;
  for (int hh = 0; hh < 64; ++hh) mx = fmaxf(mx, col[hh << 6]);
  float s = 0.f;
  for (int hh = 0; hh < 64; ++hh) s += __expf(col[hh << 6] - mx);
  const float inv = 1.f / s;
  float* ocol = wsm + (size_t)bch * HWN + w;
  for (int hh = 0; hh < 64; ++hh) ocol[hh << 6] = __expf(col[hh << 6] - mx) * inv;
}

// ---------------------------------------------------------------------------
// K5: CARAFE reassembly (bandwidth bound: 134 MB output). Weight tile
//     (36 x 64) staged by one 2D TDM load; shared by 4 channel groups.
// ---------------------------------------------------------------------------
__global__ __launch_bounds__(256)
void k5_reassemble(const float* __restrict__ x, const float* __restrict__ wsm,
                   float* __restrict__ out) {
  __shared__ float wl[ENCCH][WWD];   // [j][w], 9 KB
  const int tid  = threadIdx.x;
  const int w    = tid & 63;
  const int csub = tid >> 6;         // 0..3
  const int h = blockIdx.y;
  const int b = blockIdx.z;
  const int c = blockIdx.x * 4 + csub;

#if HAVE_TDM
  if (tid < 32) {
    // 2D tile: X=w (64 contiguous), Y=j (36 rows, stride 4096 elements)
    tdm_load_to_lds_f32((uint)(size_t)(void*)wl,
                        wsm + (size_t)b * ENCCH * HWN + (h << 6),
                        64u, (uint)ENCCH, 0u, 64u, (uint)ENCCH, 0u,
                        (unsigned long long)HWN, 0ull);
    __builtin_amdgcn_s_wait_tensorcnt(0);
  }
#else
  if (csub == 0)
    for (int j = 0; j < ENCCH; ++j)
      wl[j][w] = wsm[((size_t)b * ENCCH + j) * HWN + (h << 6) + w];
#endif
  __syncthreads();

  const float* xb = x + ((size_t)b * CI + c) * HWN;
  float a0 = 0.f, a1 = 0.f, a2 = 0.f, a3 = 0.f;
#pragma unroll
  for (int k2 = 0; k2 < 9; ++k2) {
    const int dy = k2 / 3 - 1;
    const int dx = k2 - 3 * (k2 / 3) - 1;
    const int hh = min(max(h + dy, 0), 63);       // replicate pad
    const int ww = min(max(w + dx, 0), 63);
    const float xv = xb[(hh << 6) + ww];
    a0 += xv * wl[0 * 9 + k2][w];
    a1 += xv * wl[1 * 9 + k2][w];
    a2 += xv * wl[2 * 9 + k2][w];
    a3 += xv * wl[3 * 9 + k2][w];
  }
  // out[b,c, 2h + s2/2, (s2%2)*64 + w]  (matches reference reshape/transpose)
  float* ob = out + ((size_t)b * CI + c) * (size_t)(128 * 128);
  ob[(2 * h + 0) * 128 + 0 * 64 + w] = a0;
  ob[(2 * h + 0) * 128 + 1 * 64 + w] = a1;
  ob[(2 * h + 1) * 128 + 0 * 64 + w] = a2;
  ob[(2 * h + 1) * 128 + 1 * 64 + w] = a3;
}

// ---------------------------------------------------------------------------
extern "C" void kernel_launch(void* const* d_in, const int* in_sizes, int n_in,
                              void* d_out, int out_size, void* d_ws, size_t ws_size,
                              hipStream_t stream) {
  const float* x     = (const float*)d_in[0];
  const float* w1    = (const float*)d_in[1];
  const float* b1    = (const float*)d_in[2];
  const float* gamma = (const float*)d_in[3];
  const float* beta  = (const float*)d_in[4];
  const float* w2    = (const float*)d_in[5];
  const float* b2    = (const float*)d_in[6];
  float* out = (float*)d_out;
  float* ws  = (float*)d_ws;

  float* comp = ws + OFS_COMP;
  float* enc  = ws + OFS_ENC;
  float* wsm  = ws + OFS_WSM;
  float* stat = ws + OFS_STAT;          // sum/sumsq then scale/shift
  float* w2p  = ws + OFS_W2P;

  (void)hipMemsetAsync(stat, 0, 128 * sizeof(float), stream);

  k1_compress<<<NPOS / 128, 256, 0, stream>>>(x, w1, b1, comp, stat);
  k2_stats<<<1, 64, 0, stream>>>(stat, gamma, beta, stat + 128);
  k2b_norm<<<(BB * CCH * HWN) / 256, 256, 0, stream>>>(comp, stat + 128);
  k2w_transpose<<<(KE * 48) / 256, 256, 0, stream>>>(w2, w2p);
  k3_encoder<<<NPOS / 128, 256, 0, stream>>>(comp, w2p, b2, enc);
  k4_softmax<<<(BB * ENCCH * WWD + 255) / 256, 256, 0, stream>>>(enc, wsm);
  k5_reassemble<<<dim3(CI / 4, HH, BB), 256, 0, stream>>>(x, wsm, out);
}